// RecurrentGCN_43198781063868
// MI455X (gfx1250) — compile-verified
//
#include <hip/hip_runtime.h>
#include <math.h>

// ---------------------------------------------------------------------------
// RecurrentGCN / TGCN, output = y[2] only.
// Exploits: GCN linearity (one weighted-sum vector v feeds all gate convs),
// H0 == 0 (drops R/cr entirely, halves each gate linear), and the fact that
// only node 2's row is needed. Remaining dense math = five 64x64 f32 matvecs
// done with V_WMMA_F32_16X16X4_F32 on a single wave32.
// ---------------------------------------------------------------------------

typedef float v2f __attribute__((ext_vector_type(2)));
typedef float v8f __attribute__((ext_vector_type(8)));

#define HD 64
#define TARGET_NODE 2

// ---------------------------------------------------------------------------
// out[0:64] = vin[0:64] @ W[64][64]   (W row-major, K-major rows)
// D = A x B with A(16x4): row 0 = 4-wide chunk of vin, rows 1..15 = 0.
// Per CDNA5 ISA 7.12.2:
//   A: lane L (L<16): M=L, vgpr0=K0, vgpr1=K1 ; lanes 16-31: K2/K3
//   B: vgpr0: lanes0-15 = row K0 (N=lane), lanes16-31 = row K2 ; vgpr1: K1/K3
//   D: vgpr0: lanes0-15 = row M=0 (N=lane)  -> our result row
// A-fragment is built with an UNCONDITIONAL ds_load_b64 followed by
// v_cndmask selects, so EXEC stays all-ones through the WMMA stream
// (no saveexec-predicated loads).
// ---------------------------------------------------------------------------
__device__ __forceinline__ void matvec64_wmma(const float* __restrict__ W,
                                              const float* vin,   // LDS, 64 f32
                                              float* vout,        // LDS, 64 f32
                                              int lane) {
  const int  l    = lane & 15;             // column index within half-wave
  const int  k0   = (lane & 16) ? 2 : 0;   // which K pair this half-wave holds
  const bool row0 = (l == 0);              // only lane 0 / lane 16 carry A row 0
#pragma unroll
  for (int n = 0; n < 4; ++n) {            // four 16-wide N tiles
    v8f acc = {};
#pragma unroll
    for (int k = 0; k < 16; ++k) {         // K = 64 in steps of 4
      const int kb = 4 * k;
      // one ds_load_b64 for the K-pair (8B aligned: 4*(kb+k0) % 8 == 0)
      const v2f av = *(const v2f*)(vin + kb + k0);
      v2f a, b;
      a.x = row0 ? av.x : 0.0f;            // v_cndmask, no EXEC change
      a.y = row0 ? av.y : 0.0f;
      b.x = W[(kb + k0)     * HD + 16 * n + l];   // B 4x16 tile of W
      b.y = W[(kb + k0 + 1) * HD + 16 * n + l];
      acc = __builtin_amdgcn_wmma_f32_16x16x4_f32(
          /*neg_a=*/false, a, /*neg_b=*/false, b,
          /*c_mod=*/(short)0, acc, /*reuse_a=*/false, /*reuse_b=*/false);
    }
    if (lane < 16) vout[16 * n + l] = acc[0];     // D row 0
  }
}

// ---------------------------------------------------------------------------
// Pass over all edges: deg[dst]++ (in-degree; +1 self loop added later) and
// compact the sources of edges into TARGET_NODE. Bandwidth-bound: 6.4 MB of
// dst + L2-resident atomics into a 400 KB deg array.
// ---------------------------------------------------------------------------
__global__ void deg_and_gather_kernel(const int* __restrict__ dst,
                                      const int* __restrict__ src,
                                      int* __restrict__ deg,
                                      int* __restrict__ count,
                                      int* __restrict__ list,
                                      int cap, int E) {
  int e = blockIdx.x * blockDim.x + threadIdx.x;
  if (e >= E) return;
  int d = dst[e];
  atomicAdd(&deg[d], 1);
  if (d == TARGET_NODE) {
    int i = atomicAdd(count, 1);
    if (i < cap) list[i] = src[e];
  }
}

// ---------------------------------------------------------------------------
// Single-wave kernel: build v, run the collapsed TGCN cell for node 2.
// ---------------------------------------------------------------------------
__global__ __launch_bounds__(32, 1)
void tgcn_node2_kernel(const float* __restrict__ x,
                       const float* __restrict__ Wz, const float* __restrict__ bz,
                       const float* __restrict__ Wh, const float* __restrict__ bh,
                       const float* __restrict__ LzW, const float* __restrict__ Lzb,
                       const float* __restrict__ LhW, const float* __restrict__ Lhb,
                       const float* __restrict__ W1,  const float* __restrict__ b1,
                       const float* __restrict__ gamma, const float* __restrict__ beta,
                       const float* __restrict__ rmean, const float* __restrict__ rvar,
                       const float* __restrict__ W2,  const float* __restrict__ b2,
                       const int* __restrict__ deg, const int* __restrict__ count,
                       const int* __restrict__ list, int cap,
                       float* __restrict__ out) {
  __shared__ float v[HD], c1[HD], c2[HD], g1[HD], g2[HD], t1[HD], red[32];
  const int lane = threadIdx.x;

  // Warm the weight matrices (global_prefetch_b8) while we chase the edge list.
  __builtin_prefetch(Wz,  0, 3);
  __builtin_prefetch(Wh,  0, 3);
  __builtin_prefetch(LzW, 0, 3);
  __builtin_prefetch(LhW, 0, 3);
  __builtin_prefetch(W1,  0, 3);

  int K = *count; if (K > cap) K = cap;
  const float dinv2 = rsqrtf((float)deg[TARGET_NODE] + 1.0f);  // +1 self loop

  // v = dinv2^2 * x[2]  +  Σ_e dinv[src_e]*dinv2 * x[src_e]
  for (int f = lane; f < HD; f += 32) {
    float acc = dinv2 * dinv2 * x[TARGET_NODE * HD + f];
    for (int e = 0; e < K; ++e) {
      int s = list[e];
      acc += dinv2 * rsqrtf((float)deg[s] + 1.0f) * x[s * HD + f];
    }
    v[f] = acc;
  }
  __syncthreads();

  // Gate convs (cr dropped: it only multiplies H0 == 0)
  matvec64_wmma(Wz, v, c1, lane);   // cz (pre-bias)
  matvec64_wmma(Wh, v, c2, lane);   // ch (pre-bias)
  __syncthreads();
  for (int f = lane; f < HD; f += 32) { c1[f] += bz[f]; c2[f] += bh[f]; }
  __syncthreads();

  // Gate linears: concat([c, 0]) @ L_W == c @ L_W[:64]  (top 64 rows)
  matvec64_wmma(LzW, c1, g1, lane);
  matvec64_wmma(LhW, c2, g2, lane);
  __syncthreads();

  // GRU update with H0 = 0:  h = (1 - Z) * tanh(...)
  for (int f = lane; f < HD; f += 32) {
    float Z  = 1.0f / (1.0f + expf(-(g1[f] + Lzb[f])));
    float Ht = tanhf(g2[f] + Lhb[f]);
    float h  = (1.0f - Z) * Ht;
    t1[f] = fmaxf(h, 0.0f);                       // relu
  }
  __syncthreads();

  // MLP: y @ W1 + b1, BN(eval), relu
  matvec64_wmma(W1, t1, g1, lane);
  __syncthreads();
  for (int f = lane; f < HD; f += 32) {
    float t = g1[f] + b1[f];
    t = (t - rmean[f]) * rsqrtf(rvar[f] + 1e-5f) * gamma[f] + beta[f];
    t1[f] = fmaxf(t, 0.0f);
  }
  __syncthreads();

  // Final 64-dot with W2 + b2
  red[lane] = t1[lane] * W2[lane] + t1[lane + 32] * W2[lane + 32];
  __syncthreads();
  if (lane == 0) {
    float s = b2[0];
    for (int i = 0; i < 32; ++i) s += red[i];
    out[0] = s;
  }
}

extern "C" void kernel_launch(void* const* d_in, const int* in_sizes, int n_in,
                              void* d_out, int out_size, void* d_ws, size_t ws_size,
                              hipStream_t stream) {
  const float* x     = (const float*)d_in[0];
  const float* Wz    = (const float*)d_in[1];
  const float* bz    = (const float*)d_in[2];
  // d_in[3] Wr, d_in[4] br : dead (R only multiplies H0 == 0)
  const float* Wh    = (const float*)d_in[5];
  const float* bh    = (const float*)d_in[6];
  const float* LzW   = (const float*)d_in[7];
  const float* Lzb   = (const float*)d_in[8];
  // d_in[9] Lr_W, d_in[10] Lr_b : dead
  const float* LhW   = (const float*)d_in[11];
  const float* Lhb   = (const float*)d_in[12];
  const float* W1    = (const float*)d_in[13];
  const float* b1    = (const float*)d_in[14];
  const float* gamma = (const float*)d_in[15];
  const float* beta  = (const float*)d_in[16];
  const float* rmean = (const float*)d_in[17];
  const float* rvar  = (const float*)d_in[18];
  const float* W2    = (const float*)d_in[19];
  const float* b2    = (const float*)d_in[20];
  const int*   src   = (const int*)d_in[21];
  const int*   dst   = (const int*)d_in[22];

  const int N = in_sizes[0] / HD;       // 100000
  const int E = in_sizes[21];           // 1600000

  // Workspace layout: [0..15] count (int at 0), [16 ..] deg[N], then list[].
  int* count = (int*)d_ws;
  int* deg   = (int*)((char*)d_ws + 16);
  int* list  = deg + N;
  size_t used = 16 + (size_t)N * sizeof(int);
  long long capll = (ws_size > used) ? (long long)((ws_size - used) / sizeof(int)) : 0;
  if (capll > E) capll = E;
  int cap = (int)capll;

  size_t zbytes = used <= ws_size ? used : ws_size;
  hipMemsetAsync(d_ws, 0, zbytes, stream);

  deg_and_gather_kernel<<<(E + 255) / 256, 256, 0, stream>>>(
      dst, src, deg, count, list, cap, E);

  tgcn_node2_kernel<<<1, 32, 0, stream>>>(
      x, Wz, bz, Wh, bh, LzW, Lzb, LhW, Lhb, W1, b1,
      gamma, beta, rmean, rvar, W2, b2,
      deg, count, list, cap, (float*)d_out);
}